// CTScene_47966194762198
// MI455X (gfx1250) — compile-verified
//
#include <hip/hip_runtime.h>
#include <math.h>

typedef unsigned int u32;
typedef __attribute__((ext_vector_type(4))) u32 u32x4;
typedef __attribute__((ext_vector_type(8))) int i32x8;
typedef __attribute__((ext_vector_type(4))) int i32x4;

#define KADJ      16
#define MAX_STEPS 128
#define T_MAX_F   100.0f    // 4.0 * INIT_SCALE(25)
#define EPS_F     1e-8f

// ---------------------------------------------------------------------------
// Kernel 1: precompute packed point table {x, y, z, 0.5*|p|^2} and
// sigma = softplus(10*density)/10 into workspace (contiguous: packed|sigma).
// ---------------------------------------------------------------------------
__global__ void CTScene_setup_kernel(const float* __restrict__ pts,
                                     const float* __restrict__ density,
                                     float4* __restrict__ packed,
                                     float* __restrict__ sigma,
                                     int n_points) {
    int i = blockIdx.x * blockDim.x + threadIdx.x;
    if (i >= n_points) return;
    float x = pts[3 * i + 0];
    float y = pts[3 * i + 1];
    float z = pts[3 * i + 2];
    float4 v;
    v.x = x; v.y = y; v.z = z;
    v.w = 0.5f * (x * x + y * y + z * z);
    packed[i] = v;
    // numerically-stable softplus(10*d)/10  (ACT_SCALE == 1)
    float a = 10.0f * density[i];
    float sp = fmaxf(a, 0.0f) + log1pf(expf(-fabsf(a)));
    sigma[i] = sp * 0.1f;
}

// ---------------------------------------------------------------------------
// Kernel 2: per-ray traversal. Point table + sigma staged into LDS with ONE
// Tensor Data Mover DMA (160 KB) per block, then wave32 traversal with LDS
// gathers for neighbor points/sigma and b128 global loads for adjacency.
// ---------------------------------------------------------------------------
__global__ __launch_bounds__(256) void CTScene_trace_kernel(
    const float* __restrict__ rays,
    const int*   __restrict__ adj,       // n_points * 16 neighbor ids
    const int*   __restrict__ start,
    const float4* __restrict__ packed,   // global table: packed[n] | sigma[n]
    float* __restrict__ out,
    int n_rays, int n_points)
{
    extern __shared__ float4 s_pts[];    // n_points*16 B points + n_points*4 B sigma
    float* s_sigma = reinterpret_cast<float*>(s_pts + n_points);

    // ---- TDM DMA: global packed[]|sigma[] -> LDS (first wave issues it) ----
    if (threadIdx.x < 32u) {
        u32 lds_off = (u32)(uintptr_t)(void*)&s_pts[0];   // flat low 32b == LDS offset
        unsigned long long ga = (unsigned long long)(uintptr_t)packed;
        u32 nelem8 = ((u32)n_points * 20u) >> 3;          // total bytes / 8

        u32x4 g0;
        g0[0] = 1u;                                        // count=1 valid descriptor
        g0[1] = lds_off;                                   // lds_addr
        g0[2] = (u32)(ga & 0xffffffffull);                 // global_addr[31:0]
        g0[3] = (u32)((ga >> 32) & 0x1ffffffull)           // global_addr[56:32]
              | (2u << 30);                                // type = 2 ("image")

        i32x8 g1;
        g1[0] = (int)(3u << 16);                           // data_size = 8B; mask=0
        g1[1] = (int)((nelem8 & 0xffffu) << 16);           // tensor_dim0[15:0]
        g1[2] = (int)((nelem8 >> 16) | (1u << 16));        // tensor_dim0[31:16]; tensor_dim1=1
        g1[3] = (int)((nelem8 & 0xffffu) << 16);           // tile_dim0
        g1[4] = 0;                                         // tile_dim1=0, tile_dim2=0
        g1[5] = (int)nelem8;                               // tensor_dim0_stride[31:0]
        g1[6] = 0;                                         // stride high / dim1_stride
        g1[7] = 0;

        i32x4 g2 = {0, 0, 0, 0};                           // groups 2/3 unused (<=2D)
        i32x4 g3 = {0, 0, 0, 0};
        i32x8 g4 = {0, 0, 0, 0, 0, 0, 0, 0};               // extended group (unused)
        __builtin_amdgcn_tensor_load_to_lds(g0, g1, g2, g3, g4, 0);
        __builtin_amdgcn_s_wait_tensorcnt(0);
    }
    __syncthreads();

    int ray = blockIdx.x * blockDim.x + threadIdx.x;
    if (ray >= n_rays) return;

    const float ox = rays[6 * ray + 0];
    const float oy = rays[6 * ray + 1];
    const float oz = rays[6 * ray + 2];
    const float dx = rays[6 * ray + 3];
    const float dy = rays[6 * ray + 4];
    const float dz = rays[6 * ray + 5];

    int    cur = start[ray];
    float4 p   = s_pts[cur];        // carried current cell {x,y,z,0.5|p|^2}
    float  sig = s_sigma[cur];
    float  t = 0.0f, acc = 0.0f;
    bool   active = true;

    for (int step = 0; step < MAX_STEPS; ++step) {
        if (!__any(active ? 1 : 0)) break;   // whole wave dead -> exit

        const int4* arow = reinterpret_cast<const int4*>(adj) + (size_t)cur * 4;
        int4 n0 = arow[0];
        int4 n1 = arow[1];
        int4 n2 = arow[2];
        int4 n3 = arow[3];

        float  best_t  = __builtin_inff();
        int    best_id = cur;
        float4 best_q  = p;
        const float tgate = t + 1e-6f;

        auto consider = [&](int id) {
            float4 q = s_pts[id];                    // ds_load_b128
            float ddx = q.x - p.x, ddy = q.y - p.y, ddz = q.z - p.z;
            float den = ddx * dx + ddy * dy + ddz * dz;
            float num = (q.w - p.w) - (ddx * ox + ddy * oy + ddz * oz);
            float tq  = (den > EPS_F) ? (num / den) : __builtin_inff();
            tq = (tq > tgate) ? tq : __builtin_inff();
            if (tq < best_t) { best_t = tq; best_id = id; best_q = q; }
        };
        consider(n0.x); consider(n0.y); consider(n0.z); consider(n0.w);
        consider(n1.x); consider(n1.y); consider(n1.z); consider(n1.w);
        consider(n2.x); consider(n2.y); consider(n2.z); consider(n2.w);
        consider(n3.x); consider(n3.y); consider(n3.z); consider(n3.w);

        bool has_exit = best_t < T_MAX_F;
        if (active) {
            float t_new = has_exit ? best_t : T_MAX_F;
            acc += fmaxf(t_new - t, 0.0f) * sig;
            t = t_new;
            if (has_exit) {
                cur = best_id;
                p   = best_q;          // new cell == chosen neighbor (no reload)
                sig = s_sigma[cur];    // ds_load_b32
            }
            active = has_exit;
        }
    }

    out[ray] = acc;
}

// ---------------------------------------------------------------------------
// Launch
// ---------------------------------------------------------------------------
extern "C" void kernel_launch(void* const* d_in, const int* in_sizes, int n_in,
                              void* d_out, int out_size, void* d_ws, size_t ws_size,
                              hipStream_t stream) {
    // setup_inputs() order:
    //   0: points  (N,3) f32      1: density (N,1) f32   2: rays (R,6) f32
    //   3: point_adjacency (N*16) i32   4: offsets (N+1) i32   5: start_point (R) i32
    const float* pts     = (const float*)d_in[0];
    const float* density = (const float*)d_in[1];
    const float* rays    = (const float*)d_in[2];
    const int*   adj     = (const int*)d_in[3];
    const int*   start   = (const int*)d_in[5];

    int n_points = in_sizes[0] / 3;
    int n_rays   = out_size;

    // Workspace layout (contiguous so one TDM DMA covers both):
    //   [ float4 packed[n_points] | float sigma[n_points] ]
    float4* packed = (float4*)d_ws;
    float*  sigma  = (float*)((char*)d_ws + (size_t)n_points * sizeof(float4));

    CTScene_setup_kernel<<<(n_points + 255) / 256, 256, 0, stream>>>(
        pts, density, packed, sigma, n_points);

    size_t shmem = (size_t)n_points * 20u;   // 160 KB for N=8192
    CTScene_trace_kernel<<<(n_rays + 255) / 256, 256, shmem, stream>>>(
        rays, adj, start, packed, (float*)d_out, n_rays, n_points);
}